// Net_33432025432566
// MI455X (gfx1250) — compile-verified
//
#include <hip/hip_runtime.h>

typedef float v2f __attribute__((ext_vector_type(2)));
typedef float v8f __attribute__((ext_vector_type(8)));

#define NNODES 100000      // divisible by 16 -> no row guards needed in GEMM stores
#define NEDGES 1600000

// ---------------- degree / normalization ----------------

__global__ void deg_init_k(float* deg, int n) {
    int i = blockIdx.x * blockDim.x + threadIdx.x;
    if (i < n) deg[i] = 1.0f;  // self-loop contributes 1 to in-degree
}

__global__ void deg_accum_k(const int* __restrict__ dst, float* deg, int e) {
    int i = blockIdx.x * blockDim.x + threadIdx.x;
    if (i < e) atomicAdd(&deg[dst[i]], 1.0f);
}

__global__ void deg_finish_k(float* deg, int n) {
    int i = blockIdx.x * blockDim.x + threadIdx.x;
    if (i < n) {
        float d = deg[i];
        deg[i] = (d > 0.0f) ? rsqrtf(fmaxf(d, 1.0f)) : 0.0f;
    }
}

// ---------------- WMMA f32 GEMM: Y[M x NOUT] = X[M x K] @ W[K x NOUT] (+bias) --------------
// One wave32 per 16x16 output tile; K stepped by 4 via V_WMMA_F32_16X16X4_F32.
// A layout : lane l (m = l&15, h = l>>4) holds A[m][2h+0], A[m][2h+1]
// B layout : lane l (n = l&15, h = l>>4) holds B[2h+0][n], B[2h+1][n]
// C/D      : lane l, vgpr v -> (row = v + 8h, col = l&15)
// Out-of-range cols are CLAMPED (not predicated): their products land only in
// D cols that are never stored, so the hot loop has no EXEC manipulation.
// M is compile-time (= NNODES, multiple of 16) so row guards vanish entirely.

template<int K, int NOUT, bool BIAS>
__global__ __launch_bounds__(32) void gemm_wmma_f32_k(
    const float* __restrict__ X, const float* __restrict__ W,
    const float* __restrict__ bias, float* __restrict__ Y)
{
    constexpr int TILESN = (NOUT + 15) / 16;
    static_assert((NNODES & 15) == 0, "row tiles are exact");
    int tile = blockIdx.x;
    int tm = tile / TILESN;
    int tn = tile - tm * TILESN;
    int lane = threadIdx.x;
    int half = lane >> 4;
    int l16  = lane & 15;
    int row  = tm * 16 + l16;            // A-matrix row for this lane (always < NNODES)
    int col  = tn * 16 + l16;            // B/C/D column for this lane
    int colc = min(col, NOUT - 1);       // clamp: garbage only pollutes unstored cols

    const float* Arow = X + (size_t)row * K;
    const float* Bcol = W + colc;

    v8f acc = {};
    constexpr int K4 = K & ~3;

#pragma unroll 4
    for (int k0 = 0; k0 < K4; k0 += 4) {
        int ka = k0 + 2 * half;
        v2f a; a.x = Arow[ka];
               a.y = Arow[ka + 1];
        v2f b; b.x = Bcol[(size_t)ka * NOUT];
               b.y = Bcol[(size_t)(ka + 1) * NOUT];
        // (neg_a, A, neg_b, B, c_mod, C, reuse_a, reuse_b)
        acc = __builtin_amdgcn_wmma_f32_16x16x4_f32(
            false, a, false, b, (short)0, acc, false, false);
    }

    if constexpr (K4 < K) {              // K tail (K=85, 50, 30 here), branch-free selects
        int ka  = K4 + 2 * half;
        bool v0 = (ka     < K);
        bool v1 = (ka + 1 < K);
        int ka0 = v0 ? ka     : K - 1;
        int ka1 = v1 ? ka + 1 : K - 1;
        float a0 = Arow[ka0], a1 = Arow[ka1];
        float b0 = Bcol[(size_t)ka0 * NOUT], b1 = Bcol[(size_t)ka1 * NOUT];
        v2f a; a.x = v0 ? a0 : 0.0f; a.y = v1 ? a1 : 0.0f;
        v2f b; b.x = v0 ? b0 : 0.0f; b.y = v1 ? b1 : 0.0f;
        acc = __builtin_amdgcn_wmma_f32_16x16x4_f32(
            false, a, false, b, (short)0, acc, false, false);
    }

    if (col < NOUT) {
        float bv = 0.0f;
        if constexpr (BIAS) bv = bias[col];
        float* yp = Y + (size_t)(tm * 16 + 8 * half) * NOUT + col;
#pragma unroll
        for (int v = 0; v < 8; ++v)      // rows tm*16 + 8h + v, all in range
            yp[(size_t)v * NOUT] = acc[v] + bv;
    }
}

// ---------------- aggregation ----------------

__global__ void agg_init_k(float* agg, const float* __restrict__ b, long long n, int F) {
    long long i = (long long)blockIdx.x * blockDim.x + threadIdx.x;
    if (i < n) agg[i] = b[(int)(i % F)];
}

// one wave per edge (real edges [0,E), self-loops [E, E+N)); lanes stride features
__global__ void scatter_k(const float* __restrict__ H,
                          const int* __restrict__ src,
                          const int* __restrict__ dst,
                          const float* __restrict__ dinv,
                          float* agg, int E, int Nn, int F)
{
    int warp = (int)((blockIdx.x * (long long)blockDim.x + threadIdx.x) >> 5);
    int lane = threadIdx.x & 31;
    int total = E + Nn;
    if (warp >= total) return;
    int s, d;
    if (warp < E) { s = src[warp]; d = dst[warp]; }
    else          { s = d = warp - E; }
    float nm = dinv[s] * dinv[d];
    const float* hrow = H + (size_t)s * F;
    float* arow = agg + (size_t)d * F;
    for (int f = lane; f < F; f += 32)
        atomicAdd(&arow[f], hrow[f] * nm);
}

__global__ void relu_k(float* a, long long n) {
    long long i = (long long)blockIdx.x * blockDim.x + threadIdx.x;
    if (i < n) a[i] = fmaxf(a[i], 0.0f);
}

// ---------------- driver ----------------

static inline void agg_layer(const float* ht, const int* srcp, const int* dstp,
                             const float* dinv, float* agg, const float* bvec,
                             int fout, hipStream_t stream)
{
    long long nelem = (long long)NNODES * fout;
    agg_init_k<<<(unsigned)((nelem + 255) / 256), 256, 0, stream>>>(agg, bvec, nelem, fout);
    int warps  = NEDGES + NNODES;
    int blocks = (warps + 7) / 8;                  // 256 threads = 8 wave32s / block
    scatter_k<<<blocks, 256, 0, stream>>>(ht, srcp, dstp, dinv, agg,
                                          NEDGES, NNODES, fout);
    relu_k<<<(unsigned)((nelem + 255) / 256), 256, 0, stream>>>(agg, nelem);
}

extern "C" void kernel_launch(void* const* d_in, const int* in_sizes, int n_in,
                              void* d_out, int out_size, void* d_ws, size_t ws_size,
                              hipStream_t stream) {
    (void)in_sizes; (void)n_in; (void)out_size; (void)ws_size;

    const float* x    = (const float*)d_in[0];
    const int*   eidx = (const int*)d_in[1];          // [2, E] -> src row 0, dst row 1
    const int*   srcp = eidx;
    const int*   dstp = eidx + NEDGES;
    const float* W1 = (const float*)d_in[2];  const float* b1 = (const float*)d_in[3];
    const float* W2 = (const float*)d_in[4];  const float* b2 = (const float*)d_in[5];
    const float* W3 = (const float*)d_in[6];  const float* b3 = (const float*)d_in[7];
    const float* W4 = (const float*)d_in[8];  const float* b4 = (const float*)d_in[9];
    const float* W5 = (const float*)d_in[10]; const float* b5 = (const float*)d_in[11];
    const float* Wl = (const float*)d_in[12]; const float* bl = (const float*)d_in[13];
    float* out = (float*)d_out;

    // workspace: bufA (N*85) | ht (N*85) | dinv (N)
    float* bufA = (float*)d_ws;
    float* ht   = bufA + (size_t)NNODES * 85;
    float* dinv = ht   + (size_t)NNODES * 85;

    // --- symmetric normalization (shared by all layers) ---
    deg_init_k  <<<(NNODES + 255) / 256, 256, 0, stream>>>(dinv, NNODES);
    deg_accum_k <<<(NEDGES + 255) / 256, 256, 0, stream>>>(dstp, dinv, NEDGES);
    deg_finish_k<<<(NNODES + 255) / 256, 256, 0, stream>>>(dinv, NNODES);

    const int TM = NNODES / 16;   // 6250 (exact)

    // layer 1: 128 -> 85
    gemm_wmma_f32_k<128, 85, false><<<TM * 6, 32, 0, stream>>>(x,    W1, nullptr, ht);
    agg_layer(ht, srcp, dstp, dinv, bufA, b1, 85, stream);
    // layer 2: 85 -> 80
    gemm_wmma_f32_k< 85, 80, false><<<TM * 5, 32, 0, stream>>>(bufA, W2, nullptr, ht);
    agg_layer(ht, srcp, dstp, dinv, bufA, b2, 80, stream);
    // layer 3: 80 -> 50
    gemm_wmma_f32_k< 80, 50, false><<<TM * 4, 32, 0, stream>>>(bufA, W3, nullptr, ht);
    agg_layer(ht, srcp, dstp, dinv, bufA, b3, 50, stream);
    // layer 4: 50 -> 30
    gemm_wmma_f32_k< 50, 30, false><<<TM * 2, 32, 0, stream>>>(bufA, W4, nullptr, ht);
    agg_layer(ht, srcp, dstp, dinv, bufA, b4, 30, stream);
    // layer 5: 30 -> 20
    gemm_wmma_f32_k< 30, 20, false><<<TM * 2, 32, 0, stream>>>(bufA, W5, nullptr, ht);
    agg_layer(ht, srcp, dstp, dinv, bufA, b5, 20, stream);

    // classifier: out = h5 @ Wl + bl  (no aggregation / relu)
    gemm_wmma_f32_k< 20,  4, true><<<TM * 1, 32, 0, stream>>>(bufA, Wl, bl, out);
}